// SymQuantizer_31112743092741
// MI455X (gfx1250) — compile-verified
//
#include <hip/hip_runtime.h>
#include <hip/hip_bf16.h>

#define COLS 4096
#define THREADS 256
#define MAXQ 7.0f

typedef float v4f __attribute__((ext_vector_type(4)));

// Per-row int4 symmetric quantization.
// One workgroup (8 wave32s) per row. Row staged into LDS with CDNA5 async
// global->LDS b128 copies (ASYNCcnt, non-temporal load hint), consumed from
// LDS into registers, reduced (wave32 shfl + LDS partials), quantized, and
// streamed out with non-temporal b128 stores.
__global__ __launch_bounds__(THREADS) void sym_quant_kernel(
    const float* __restrict__ x,
    float* __restrict__ q,
    float* __restrict__ scales) {
  __shared__ float tile[COLS];     // 16 KiB row buffer
  __shared__ float red[8];         // per-wave partial maxima / scale broadcast

  const unsigned tid = threadIdx.x;
  const unsigned row = blockIdx.x;
  const unsigned row_byte = row * (COLS * 4u);   // < 2^27, fits i32 voffset

  // ---- Stage row into LDS: 4 x b128 per lane, async, read-once hint ----
  const unsigned lds_base = (unsigned)(size_t)(&tile[0]);
#pragma unroll
  for (int i = 0; i < 4; ++i) {
    const unsigned elem  = (unsigned)i * 1024u + tid * 4u;  // float index
    const unsigned boff  = elem * 4u;                       // byte offset in row
    const unsigned ldsa  = lds_base + boff;
    const unsigned voff  = row_byte + boff;
    asm volatile("global_load_async_to_lds_b128 %0, %1, %2 th:TH_LOAD_NT"
                 :
                 : "v"(ldsa), "v"(voff), "s"(x)
                 : "memory");
  }
  asm volatile("s_wait_asynccnt 0x0" ::: "memory");
  __syncthreads();

  // ---- Pull 16 floats into registers, local abs-max ----
  float v[16];
  float amax = 0.0f;
#pragma unroll
  for (int i = 0; i < 4; ++i) {
    const v4f f = *(const v4f*)&tile[(unsigned)i * 1024u + tid * 4u];
    v[i * 4 + 0] = f.x;
    v[i * 4 + 1] = f.y;
    v[i * 4 + 2] = f.z;
    v[i * 4 + 3] = f.w;
    amax = fmaxf(amax, fabsf(f.x));
    amax = fmaxf(amax, fabsf(f.y));
    amax = fmaxf(amax, fabsf(f.z));
    amax = fmaxf(amax, fabsf(f.w));
  }

  // ---- wave32 tree reduce, then cross-wave via LDS ----
#pragma unroll
  for (int off = 16; off > 0; off >>= 1)
    amax = fmaxf(amax, __shfl_xor(amax, off, 32));
  if ((tid & 31u) == 0u) red[tid >> 5] = amax;
  __syncthreads();

  if (tid == 0) {
    float m = red[0];
#pragma unroll
    for (int i = 1; i < 8; ++i) m = fmaxf(m, red[i]);
    // scale = fp16(amax / 7) * clip_ratio(=1), kept as the fp16-rounded value
    const float s = (float)(_Float16)(m / MAXQ);
    red[0] = s;
    scales[row] = s;
  }
  __syncthreads();

  const float s   = red[0];
  const float inv = 1.0f / s;

  // ---- Quantize from registers, streaming (non-temporal) b128 stores ----
  float* __restrict__ qrow = q + (size_t)row * COLS;
#pragma unroll
  for (int i = 0; i < 4; ++i) {
    v4f o;
    o.x = fminf(fmaxf(nearbyintf(v[i * 4 + 0] * inv), -(MAXQ + 1.0f)), MAXQ);
    o.y = fminf(fmaxf(nearbyintf(v[i * 4 + 1] * inv), -(MAXQ + 1.0f)), MAXQ);
    o.z = fminf(fmaxf(nearbyintf(v[i * 4 + 2] * inv), -(MAXQ + 1.0f)), MAXQ);
    o.w = fminf(fmaxf(nearbyintf(v[i * 4 + 3] * inv), -(MAXQ + 1.0f)), MAXQ);
    __builtin_nontemporal_store(o, (v4f*)&qrow[(unsigned)i * 1024u + tid * 4u]);
  }
}

extern "C" void kernel_launch(void* const* d_in, const int* in_sizes, int n_in,
                              void* d_out, int out_size, void* d_ws, size_t ws_size,
                              hipStream_t stream) {
  const float* x = (const float*)d_in[0];
  const int n    = in_sizes[0];          // 8192 * 4096
  const int rows = n / COLS;             // 8192

  float* q      = (float*)d_out;         // q: rows*COLS floats, then scales
  float* scales = (float*)d_out + (size_t)rows * COLS;

  sym_quant_kernel<<<rows, THREADS, 0, stream>>>(x, q, scales);
}